// SuperpixelTokenization_65824668778742
// MI455X (gfx1250) — compile-verified
//
#include <hip/hip_runtime.h>

// Problem constants (fixed by the reference setup)
#define B_    8
#define F_    64
#define H_    512
#define W_    512
#define P_    (H_ * W_)          // 262144 pixels per image
#define NSEG  2048
#define FC    16                 // feature channels per accumulate block
#define NFC   (F_ / FC)          // 4 feature chunks
#define SCH   16                 // pixel chunks per (batch, feature-chunk)
#define PC    (P_ / SCH)         // 16384 pixels per block
#define NT    512                // threads per accumulate block
#define TILES (PC / NT)          // 32 tiles of NT pixels
#define ACC_STRIDE 2056          // 2048 + 8 pad: spreads random seg ids over all 64 LDS banks

// Low 32 bits of a flat pointer into LDS == DS byte offset (LDS aperture layout).
__device__ __forceinline__ unsigned lds_offset(const void* p) {
  return (unsigned)(unsigned long long)p;
}

// CDNA5 async global->LDS copy (GVS addressing: saddr base + per-lane byte offset).
// Tracked by ASYNCcnt; completion enforced with s_wait_asynccnt.
__device__ __forceinline__ void async_load_b32(const int* sbase, unsigned vbyteoff,
                                               unsigned ldsbyte) {
  asm volatile("global_load_async_to_lds_b32 %0, %1, %2"
               :: "v"(ldsbyte), "v"(vbyteoff), "s"(sbase)
               : "memory");
}

// Guaranteed-native no-return FP32 global atomic add (TH=0 -> non-returning):
// avoids any chance of a CAS-loop expansion of atomicrmw fadd.
__device__ __forceinline__ void gatomic_add_f32(float* sbase, unsigned vbyteoff, float v) {
  asm volatile("global_atomic_add_f32 %0, %1, %2"
               :: "v"(vbyteoff), "v"(v), "s"(sbase)
               : "memory");
}

// Core pass: LDS-privatized scatter-sum. (First in file so the disasm snippet shows it.)
// Block = (batch b, feature chunk fc, pixel chunk sc). All 2048 segments x 16
// features accumulate in LDS via ds_add_f32; one global atomic per (seg,feat)
// per block at flush time (8x fewer global atomics than the naive kernel).
__global__ void __launch_bounds__(NT) accumulate_kernel(
    const float* __restrict__ feats, const int* __restrict__ seg,
    float* __restrict__ sums, float* __restrict__ cnt,
    float* __restrict__ cxs, float* __restrict__ cys) {
  __shared__ __align__(16) float acc[FC * ACC_STRIDE];   // 128.5 KB  [feature j][segment s]
  __shared__ float ccnt[NSEG];
  __shared__ float ccx[NSEG];
  __shared__ float ccy[NSEG];
  __shared__ int   segbuf[2 * NT];         // double-buffered async staging

  const int tid = threadIdx.x;
  const int sc  = blockIdx.x % SCH;
  const int fc  = (blockIdx.x / SCH) % NFC;
  const int b   = blockIdx.x / (SCH * NFC);

  const int pbase = sc * PC;
  const int* segb = seg + b * P_;
  const unsigned segbuf_lds = lds_offset(segbuf);

  // Kick off async prefetch of tile 0's segment ids while we zero LDS.
  async_load_b32(segb, (unsigned)((pbase + tid) * 4), segbuf_lds + tid * 4u);

  // Vectorized LDS zeroing (ds_store_b128)
  {
    float4* acc4 = (float4*)acc;
    const float4 z4 = make_float4(0.f, 0.f, 0.f, 0.f);
    for (int i = tid; i < (FC * ACC_STRIDE) / 4; i += NT) acc4[i] = z4;
    if (fc == 0) {
      for (int i = tid; i < NSEG; i += NT) { ccnt[i] = 0.0f; ccx[i] = 0.0f; ccy[i] = 0.0f; }
    }
  }
  __syncthreads();

  const float* fb = feats + (size_t)(b * F_ + fc * FC) * P_;

  for (int t = 0; t < TILES; ++t) {
    if (t + 1 < TILES) {
      // Prefetch next tile, then wait until only it is outstanding -> tile t ready.
      async_load_b32(segb, (unsigned)((pbase + (t + 1) * NT + tid) * 4),
                     segbuf_lds + (unsigned)(((t + 1) & 1) * NT + tid) * 4u);
      asm volatile("s_wait_asynccnt 1" ::: "memory");
    } else {
      asm volatile("s_wait_asynccnt 0" ::: "memory");
    }
    const int p = pbase + t * NT + tid;
    const int s = segbuf[(t & 1) * NT + tid];

#pragma unroll
    for (int j = 0; j < FC; ++j) {
      // coalesced, single-use -> non-temporal
      float v = __builtin_nontemporal_load(fb + (size_t)j * P_ + p);
      atomicAdd(&acc[j * ACC_STRIDE + s], v);   // ds_add_f32 (no return)
    }
    if (fc == 0) {
      float xn = (float)(p & (W_ - 1)) * (1.0f / (float)(W_ - 1));
      float yn = (float)((p >> 9) & (H_ - 1)) * (1.0f / (float)(H_ - 1));
      atomicAdd(&ccnt[s], 1.0f);
      atomicAdd(&ccx[s], xn);
      atomicAdd(&ccy[s], yn);
    }
  }
  __syncthreads();

  // Flush LDS partials to global (coalesced: 16 consecutive features per segment).
  float* sums_b = sums + (size_t)b * NSEG * F_ + fc * FC;
  for (int idx = tid; idx < NSEG * FC; idx += NT) {
    int s = idx >> 4;
    int j = idx & (FC - 1);
    float v = acc[j * ACC_STRIDE + s];
    if (v != 0.0f) gatomic_add_f32(sums_b, (unsigned)((s * F_ + j) * 4), v);
  }
  if (fc == 0) {
    float* cntb = cnt + b * NSEG;
    float* cxb  = cxs + b * NSEG;
    float* cyb  = cys + b * NSEG;
    for (int s = tid; s < NSEG; s += NT) {
      float c = ccnt[s];
      if (c != 0.0f) {
        gatomic_add_f32(cntb, (unsigned)(s * 4), c);
        gatomic_add_f32(cxb,  (unsigned)(s * 4), ccx[s]);
        gatomic_add_f32(cyb,  (unsigned)(s * 4), ccy[s]);
      }
    }
  }
}

__global__ void zero_ws_kernel(unsigned* __restrict__ w, int n) {
  int i = blockIdx.x * blockDim.x + threadIdx.x;
  if (i < n) w[i] = 0u;
}

// Pass-through copy of segment_map (cast to float) fused with per-batch max reduction.
__global__ void copy_seg_max_kernel(const int* __restrict__ seg,
                                    float* __restrict__ out_seg,
                                    int* __restrict__ mx) {
  __shared__ int smax;
  if (threadIdx.x == 0) smax = 0;
  __syncthreads();
  int idx = blockIdx.x * blockDim.x + threadIdx.x;   // over B*P
  int s = __builtin_nontemporal_load(seg + idx);
  __builtin_nontemporal_store((float)s, out_seg + idx);
  atomicMax(&smax, s);
  __syncthreads();
  if (threadIdx.x == 0) atomicMax(&mx[blockIdx.x / (P_ / 256)], smax);
}

__global__ void finalize_tokens_kernel(const float* __restrict__ sums,
                                       const float* __restrict__ cnt,
                                       float* __restrict__ out_tokens) {
  int idx = blockIdx.x * blockDim.x + threadIdx.x;  // over B*NSEG*F
  float c = cnt[idx / F_];
  out_tokens[idx] = sums[idx] / fmaxf(c, 1.0f);
}

__global__ void finalize_meta_kernel(const float* __restrict__ cnt,
                                     const float* __restrict__ cxs,
                                     const float* __restrict__ cys,
                                     const int* __restrict__ mx,
                                     float* __restrict__ out_mask,
                                     float* __restrict__ out_cent) {
  int idx = blockIdx.x * blockDim.x + threadIdx.x;  // over B*NSEG
  int b = idx / NSEG;
  int s = idx % NSEG;
  float c = fmaxf(cnt[idx], 1.0f);
  out_cent[idx * 2 + 0] = cxs[idx] / c;
  out_cent[idx * 2 + 1] = cys[idx] / c;
  out_mask[idx] = (s <= mx[b]) ? 1.0f : 0.0f;
}

extern "C" void kernel_launch(void* const* d_in, const int* in_sizes, int n_in,
                              void* d_out, int out_size, void* d_ws, size_t ws_size,
                              hipStream_t stream) {
  (void)in_sizes; (void)n_in; (void)out_size; (void)ws_size;
  const float* feats = (const float*)d_in[0];
  const int*   seg   = (const int*)d_in[1];
  // d_in[2] = n_segments scalar (fixed 2048, compiled in)

  float* out        = (float*)d_out;
  float* out_tokens = out;                                   // B*NSEG*F
  float* out_seg    = out_tokens + (size_t)B_ * NSEG * F_;   // B*P
  float* out_mask   = out_seg + (size_t)B_ * P_;             // B*NSEG
  float* out_cent   = out_mask + (size_t)B_ * NSEG;          // B*NSEG*2

  float* sums = (float*)d_ws;                                // B*NSEG*F
  float* cnt  = sums + (size_t)B_ * NSEG * F_;               // B*NSEG
  float* cxs  = cnt + B_ * NSEG;                             // B*NSEG
  float* cys  = cxs + B_ * NSEG;                             // B*NSEG
  int*   mx   = (int*)(cys + B_ * NSEG);                     // B ints

  const int nzero = B_ * NSEG * F_ + 3 * B_ * NSEG + B_;
  zero_ws_kernel<<<(nzero + 255) / 256, 256, 0, stream>>>((unsigned*)d_ws, nzero);
  copy_seg_max_kernel<<<(B_ * P_) / 256, 256, 0, stream>>>(seg, out_seg, mx);
  accumulate_kernel<<<B_ * NFC * SCH, NT, 0, stream>>>(feats, seg, sums, cnt, cxs, cys);
  finalize_tokens_kernel<<<(B_ * NSEG * F_) / 256, 256, 0, stream>>>(sums, cnt, out_tokens);
  finalize_meta_kernel<<<(B_ * NSEG) / 256, 256, 0, stream>>>(cnt, cxs, cys, mx, out_mask, out_cent);
}